// RWSESPDEdgeEncoder_17377437679648
// MI455X (gfx1250) — compile-verified
//
#include <hip/hip_runtime.h>

typedef float v2f __attribute__((ext_vector_type(2)));
typedef float v8f __attribute__((ext_vector_type(8)));

#define B_   16
#define N_   128
#define M_   256
#define K_   16
#define EMB_ 64
#define E_   32768
#define ES_  65536

// ---------------------------------------------------------------------------
// counts[b] += popcount of (bvec == b)
__global__ __launch_bounds__(256) void count_kernel(const int* __restrict__ bvec,
                                                    int n, int* __restrict__ counts) {
    int i = blockIdx.x * blockDim.x + threadIdx.x;
    if (i < n) atomicAdd(&counts[bvec[i]], 1);
}

// exclusive prefix over B_ counts (single thread; B_ == 16, trivial)
__global__ void prefix_kernel(const int* __restrict__ counts, int* __restrict__ cum) {
    int s = 0;
    for (int b = 0; b < B_; ++b) { cum[b] = s; s += counts[b]; }
    cum[B_] = s;
}

// ---------------------------------------------------------------------------
// out[rows,64] = A[rows,16] * W[16,64] + bias[64]   via V_WMMA_F32_16X16X4_F32
// One wave32 per 16-row tile; 4 K-chunks x 4 column-tiles = 16 WMMA ops/wave.
__global__ __launch_bounds__(256) void rwse_gemm_wmma(const float* __restrict__ A,
                                                      const float* __restrict__ W,
                                                      const float* __restrict__ bias,
                                                      float* __restrict__ out,
                                                      int rows) {
    int wave    = blockIdx.x * (blockDim.x >> 5) + (threadIdx.x >> 5);
    int lane    = threadIdx.x & 31;
    int rowbase = wave << 4;
    if (rowbase >= rows) return;

    int lm = lane & 15;   // A: tile row      | B/C/D: column within 16-wide tile
    int hi = lane >> 4;   // selects K pair (A/B) / row half (C/D)

    // A operands: row (rowbase+lm), K = 4*kc + 2*hi + {0,1}  -> contiguous float2
    const float* arow = A + (size_t)(rowbase + lm) * K_;
    v2f a[4];
#pragma unroll
    for (int kc = 0; kc < 4; ++kc)
        a[kc] = *(const v2f*)(arow + kc * 4 + hi * 2);

#pragma unroll
    for (int ct = 0; ct < 4; ++ct) {
        int n = ct * 16 + lm;
        v8f c = {};
#pragma unroll
        for (int kc = 0; kc < 4; ++kc) {
            int kb = kc * 4 + hi * 2;
            v2f b;
            b.x = W[(size_t)kb * EMB_ + n];
            b.y = W[(size_t)(kb + 1) * EMB_ + n];
            // 8 args: (neg_a, A, neg_b, B, c_mod, C, reuse_a, reuse_b)
            c = __builtin_amdgcn_wmma_f32_16x16x4_f32(false, a[kc], false, b,
                                                      (short)0, c, false, false);
        }
        float bv = bias[n];
#pragma unroll
        for (int v = 0; v < 8; ++v) {
            int m = v + (hi << 3);
            out[(size_t)(rowbase + m) * EMB_ + n] = c[v] + bv;
        }
    }
}

// ---------------------------------------------------------------------------
// edge_attr = 0.5 * (dense[b,r,c,:] + dense[b,c,r,:])
__global__ __launch_bounds__(256) void gather_sym(const float* __restrict__ dense,
                                                  const int* __restrict__ ei,
                                                  const int* __restrict__ bvec,
                                                  const int* __restrict__ cum,
                                                  float* __restrict__ out,
                                                  int nloc, int nedges) {
    int t = blockIdx.x * blockDim.x + threadIdx.x;
    int e = t >> 6, d = t & 63;
    if (e >= nedges) return;
    int i0 = ei[e], i1 = ei[nedges + e];
    int b0 = bvec[i0];
    int r  = i0 - cum[b0];
    int c  = i1 - cum[bvec[i1]];
    size_t base = (size_t)b0 * nloc;
    float x = dense[(((base + r) * nloc) + c) * EMB_ + d];
    float y = dense[(((base + c) * nloc) + r) * EMB_ + d];
    out[(size_t)e * EMB_ + d] = 0.5f * (x + y);
}

// edge_attr = dense[b,r,c,:]
__global__ __launch_bounds__(256) void gather_dir(const float* __restrict__ dense,
                                                  const int* __restrict__ ei,
                                                  const int* __restrict__ bvec,
                                                  const int* __restrict__ cum,
                                                  float* __restrict__ out,
                                                  int nloc, int nedges) {
    int t = blockIdx.x * blockDim.x + threadIdx.x;
    int e = t >> 6, d = t & 63;
    if (e >= nedges) return;
    int i0 = ei[e], i1 = ei[nedges + e];
    int b0 = bvec[i0];
    int r  = i0 - cum[b0];
    int c  = i1 - cum[bvec[i1]];
    out[(size_t)e * EMB_ + d] =
        dense[(((size_t)b0 * nloc + r) * nloc) * EMB_ + (size_t)c * EMB_ + d];
}

// ---------------------------------------------------------------------------
// scatter-add table[len] into dense[b,r,c,:]; entries = ES edges + nnodes self-loops
__global__ __launch_bounds__(256) void spd_scatter(const int* __restrict__ spd_idx,
                                                   const int* __restrict__ spd_len,
                                                   const int* __restrict__ bvec,
                                                   const int* __restrict__ cum,
                                                   const float* __restrict__ table,
                                                   float* __restrict__ dense,
                                                   int nspd, int nnodes, int nloc) {
    int t = blockIdx.x * blockDim.x + threadIdx.x;
    int e = t >> 6, d = t & 63;
    if (e >= nspd + nnodes) return;
    int i0, i1, len;
    if (e < nspd) {
        i0  = spd_idx[e];
        i1  = spd_idx[nspd + e];
        len = spd_len[e] + 1;
    } else {                       // self-loop with fill_value 0
        i0 = i1 = e - nspd;
        len = 0;
    }
    int b0 = bvec[i0];
    int r  = i0 - cum[b0];
    int c  = i1 - cum[bvec[i1]];
    atomicAdd(&dense[(((size_t)b0 * nloc + r) * nloc + c) * EMB_ + d],
              table[(size_t)len * EMB_ + d]);
}

// ---------------------------------------------------------------------------
extern "C" void kernel_launch(void* const* d_in, const int* in_sizes, int n_in,
                              void* d_out, int out_size, void* d_ws, size_t ws_size,
                              hipStream_t stream) {
    (void)in_sizes; (void)n_in; (void)out_size; (void)ws_size;

    const float* edge_RWSE       = (const float*)d_in[0];
    const float* e2e_edge_RWSE   = (const float*)d_in[1];
    const float* W_enc           = (const float*)d_in[2];
    const float* b_enc           = (const float*)d_in[3];
    const float* W_e2e           = (const float*)d_in[4];
    const float* b_e2e           = (const float*)d_in[5];
    const float* spd_table       = (const float*)d_in[6];
    const float* e2e_spd_table   = (const float*)d_in[7];
    const int*   batch_vec       = (const int*)d_in[8];
    const int*   e_batch         = (const int*)d_in[9];
    const int*   edge_index      = (const int*)d_in[10];
    const int*   e2e_edge_index  = (const int*)d_in[11];
    const int*   spd_index       = (const int*)d_in[12];
    const int*   spd_lengths     = (const int*)d_in[13];
    const int*   e2e_spd_index   = (const int*)d_in[14];
    const int*   e2e_spd_lengths = (const int*)d_in[15];

    float* out = (float*)d_out;

    // Output layout: concatenated in reference return order.
    const size_t sz_edge  = (size_t)E_ * EMB_;              // 2,097,152
    const size_t sz_nd    = (size_t)B_ * N_ * N_ * EMB_;    // 16,777,216
    const size_t sz_ed    = (size_t)B_ * M_ * M_ * EMB_;    // 67,108,864
    const size_t off_edge_attr        = 0;
    const size_t off_dense_rwse       = off_edge_attr        + sz_edge;
    const size_t off_e2e_edge_attr    = off_dense_rwse       + sz_nd;
    const size_t off_dense_e2e        = off_e2e_edge_attr    + sz_edge;
    const size_t off_edge_attr_spd    = off_dense_e2e        + sz_ed;
    const size_t off_spd_dense        = off_edge_attr_spd    + sz_edge;
    const size_t off_e2e_edge_attr_spd= off_spd_dense        + sz_nd;
    const size_t off_e2e_spd_dense    = off_e2e_edge_attr_spd+ sz_edge;

    // Workspace: counts (2x16) + cum (2x17)
    int* counts_n = (int*)d_ws;
    int* counts_e = counts_n + 16;
    int* cum_n    = counts_n + 32;
    int* cum_e    = counts_n + 49;

    hipMemsetAsync(counts_n, 0, 32 * sizeof(int), stream);
    count_kernel<<<(B_ * N_ + 255) / 256, 256, 0, stream>>>(batch_vec, B_ * N_, counts_n);
    count_kernel<<<(B_ * M_ + 255) / 256, 256, 0, stream>>>(e_batch, B_ * M_, counts_e);
    prefix_kernel<<<1, 1, 0, stream>>>(counts_n, cum_n);
    prefix_kernel<<<1, 1, 0, stream>>>(counts_e, cum_e);

    // Dense RWSE GEMMs (WMMA f32 16x16x4), 8 waves/block, one 16-row tile/wave
    {
        int rows  = B_ * N_ * N_;              // 262144 -> 16384 waves -> 2048 blocks
        rwse_gemm_wmma<<<rows / 16 / 8, 256, 0, stream>>>(edge_RWSE, W_enc, b_enc,
                                                          out + off_dense_rwse, rows);
        rows = B_ * M_ * M_;                   // 1048576 -> 65536 waves -> 8192 blocks
        rwse_gemm_wmma<<<rows / 16 / 8, 256, 0, stream>>>(e2e_edge_RWSE, W_e2e, b_e2e,
                                                          out + off_dense_e2e, rows);
    }

    // Zero SPD dense tensors (scatter-add targets)
    hipMemsetAsync(out + off_spd_dense,     0, sz_nd * sizeof(float), stream);
    hipMemsetAsync(out + off_e2e_spd_dense, 0, sz_ed * sizeof(float), stream);

    // SPD scatter-adds (edges + self-loops)
    {
        int total = (ES_ + B_ * N_) * EMB_;
        spd_scatter<<<(total + 255) / 256, 256, 0, stream>>>(
            spd_index, spd_lengths, batch_vec, cum_n, spd_table,
            out + off_spd_dense, ES_, B_ * N_, N_);
        total = (ES_ + B_ * M_) * EMB_;
        spd_scatter<<<(total + 255) / 256, 256, 0, stream>>>(
            e2e_spd_index, e2e_spd_lengths, e_batch, cum_e, e2e_spd_table,
            out + off_e2e_spd_dense, ES_, B_ * M_, M_);
    }

    // Gathers
    {
        int total = E_ * EMB_;
        int grid  = (total + 255) / 256;
        gather_sym<<<grid, 256, 0, stream>>>(out + off_dense_rwse, edge_index,
                                             batch_vec, cum_n,
                                             out + off_edge_attr, N_, E_);
        gather_dir<<<grid, 256, 0, stream>>>(out + off_dense_e2e, e2e_edge_index,
                                             e_batch, cum_e,
                                             out + off_e2e_edge_attr, M_, E_);
        gather_dir<<<grid, 256, 0, stream>>>(out + off_spd_dense, edge_index,
                                             batch_vec, cum_n,
                                             out + off_edge_attr_spd, N_, E_);
        gather_dir<<<grid, 256, 0, stream>>>(out + off_e2e_spd_dense, e2e_edge_index,
                                             e_batch, cum_e,
                                             out + off_e2e_edge_attr_spd, M_, E_);
    }
}